// FocalLossHandMade3_19292993094178
// MI455X (gfx1250) — compile-verified
//
#include <hip/hip_runtime.h>
#include <hip/hip_bf16.h>

#define TPB 512
#define WAVES (TPB / 32)
#define TILEF (TPB * 4)            // 2048 floats (8 KB) per tile, double-buffered
#define NEG_INF (-__builtin_inff())

typedef int v4i __attribute__((ext_vector_type(4)));

#if __has_builtin(__builtin_amdgcn_global_load_async_to_lds_b128)
#define HAVE_ASYNC_LDS 1
#else
#define HAVE_ASYNC_LDS 0
#endif

#if __has_builtin(__builtin_amdgcn_s_wait_asynccnt)
#define WAIT_ASYNC(N) __builtin_amdgcn_s_wait_asynccnt(N)
#else
#define WAIT_ASYNC(N) asm volatile("s_wait_asynccnt " #N ::: "memory")
#endif

// One block per row: single streaming pass (online softmax + below-x_k max),
// staged through LDS via the CDNA5 async global->LDS engine, double-buffered.
__global__ __launch_bounds__(TPB) void focal_rows(const float* __restrict__ in,
                                                  const int* __restrict__ tgt,
                                                  float* __restrict__ row_loss,
                                                  int V) {
  __shared__ float tile[2][TILEF];
  __shared__ float red_m[WAVES], red_s[WAVES], red_b[WAVES];

  const int row = blockIdx.x;
  const int tid = threadIdx.x;
  const float* __restrict__ rowp = in + (size_t)row * (size_t)V;
  const float xk = rowp[tgt[row]];

  const int nt   = (V + TILEF - 1) / TILEF;
  const int safe = (V - 4) & ~3;   // highest 16B-aligned fully in-bounds float4 start

  // Stage tile t into LDS buffer t&1: exactly one b128 async per thread,
  // so ASYNCcnt advances by exactly 1 per wave per tile (uniform control flow).
  auto stage = [&](int t) {
    const int base  = t * TILEF + tid * 4;
    const int cbase = (base + 4 <= V) ? base : safe;  // clamp keeps it aligned + in-bounds
#if HAVE_ASYNC_LDS
    __builtin_amdgcn_global_load_async_to_lds_b128(
        (__attribute__((address_space(1))) v4i*)(rowp + cbase),
        (__attribute__((address_space(3))) v4i*)(&tile[t & 1][tid * 4]),
        0, 0);
#else
    float4 v = *(const float4*)(rowp + cbase);
    *(float4*)(&tile[t & 1][tid * 4]) = v;
#endif
  };

  stage(0);

  float m = NEG_INF;   // running row max
  float s = 0.0f;      // running sum of exp(x - m)
  float b = NEG_INF;   // max of elements strictly below xk

  for (int t = 0; t < nt; ++t) {
    if (t + 1 < nt) {
      stage(t + 1);
      WAIT_ASYNC(1);   // async loads retire in order -> tile t is in LDS
    } else {
      WAIT_ASYNC(0);   // last tile: drain
    }
    __syncthreads();   // all waves' slices of tile t visible block-wide

    const int base = t * TILEF + tid * 4;
    float4 v4 = *(const float4*)(&tile[t & 1][tid * 4]);
    const bool full = (base + 4 <= V);
    float xs[4] = {v4.x, v4.y, v4.z, v4.w};
#pragma unroll
    for (int j = 0; j < 4; ++j) {
      if (base + j < V) {
        // ragged tail chunk (at most one per row): fetch straight from global
        float x = full ? xs[j] : rowp[base + j];
        if (x > m) { s *= __expf(m - x); m = x; }   // rare rescale, coherent branch
        s += __expf(x - m);
        if (x < xk) b = fmaxf(b, x);
      }
    }
    __syncthreads();   // buffer t&1 free for reuse by stage(t+2)
  }

  // wave32 log-sum-exp merge
  auto combine = [&]() {
#pragma unroll
    for (int off = 16; off > 0; off >>= 1) {
      float m2 = __shfl_xor(m, off, 32);
      float s2 = __shfl_xor(s, off, 32);
      float b2 = __shfl_xor(b, off, 32);
      float mn = fmaxf(m, m2);
      s = s * __expf(m - mn) + s2 * __expf(m2 - mn);
      m = mn;
      b = fmaxf(b, b2);
    }
  };
  combine();

  const int wave = tid >> 5, lane = tid & 31;
  if (lane == 0) { red_m[wave] = m; red_s[wave] = s; red_b[wave] = b; }
  __syncthreads();

  if (wave == 0) {
    m = (lane < WAVES) ? red_m[lane] : NEG_INF;
    s = (lane < WAVES) ? red_s[lane] : 0.0f;
    b = (lane < WAVES) ? red_b[lane] : NEG_INF;
    combine();
    if (lane == 0) {
      const float invZ  = 1.0f / s;
      const float logpk = (xk - m) - __logf(s);
      const float pk    = __expf(xk - m) * invZ;
      const float pj    = __expf(b - m) * invZ;  // b == -inf -> exp -> 0
      const float u     = 1.0f - pk + pj;
      row_loss[row] = -(u * u) * logpk;          // gamma = 2
    }
  }
}

// Deterministic final reduction: 4096 row losses -> scalar (no float atomics).
__global__ __launch_bounds__(256) void sum_rows(const float* __restrict__ row_loss,
                                                float* __restrict__ out, int N) {
  __shared__ float red[8];
  const int tid = threadIdx.x;
  float s = 0.0f;
  for (int i = tid; i < N; i += 256) s += row_loss[i];
#pragma unroll
  for (int off = 16; off > 0; off >>= 1) s += __shfl_xor(s, off, 32);
  if ((tid & 31) == 0) red[tid >> 5] = s;
  __syncthreads();
  if (tid == 0) {
    float acc = 0.0f;
#pragma unroll
    for (int w = 0; w < 8; ++w) acc += red[w];
    out[0] = acc;
  }
}

extern "C" void kernel_launch(void* const* d_in, const int* in_sizes, int n_in,
                              void* d_out, int out_size, void* d_ws, size_t ws_size,
                              hipStream_t stream) {
  const float* in  = (const float*)d_in[0];
  const int*   tgt = (const int*)d_in[1];
  const int N = in_sizes[1];                  // 4096 rows
  const int V = in_sizes[0] / N;              // 50257 vocab
  float* rl = (float*)d_ws;                   // N floats of scratch

  focal_rows<<<N, TPB, 0, stream>>>(in, tgt, rl, V);
  sum_rows<<<1, 256, 0, stream>>>(rl, (float*)d_out, N);
}